// CTGANGenerator_6511170420789
// MI455X (gfx1250) — compile-verified
//
#include <hip/hip_runtime.h>

typedef __attribute__((ext_vector_type(2))) float v2f;
typedef __attribute__((ext_vector_type(8))) float v8f;

#define BATCH_N 65536
#define MT 128
#define NT 64
#define KT 16
#define ROWBLK (BATCH_N / MT)   // 512 row blocks
#define DATA_DIM 366
#define NSEG 40
#define TAU_INV 5.0f            // 1 / 0.2
#define BN_EPS 1e-3f

// static ragged segment layout from the CTGAN metadata (compile-time constant)
__constant__ int c_segStart[NSEG + 1] = {
    0, 1, 11, 12, 20, 21, 31, 32, 37, 38, 48, 49, 59, 60, 67, 68, 78, 79, 89, 90,
    100, 102, 107, 117, 120, 170, 174, 176, 184, 284, 290, 292, 295, 307, 312, 314,
    344, 348, 357, 359, 366};

// ---- CDNA5 async global->LDS copy (ASYNCcnt-tracked, bypasses VGPRs) -------
__device__ __forceinline__ void async_load_b32(unsigned lds_byte_addr,
                                               const void* gaddr) {
  asm volatile("global_load_async_to_lds_b32 %0, %1, off"
               :
               : "v"(lds_byte_addr), "v"(gaddr)
               : "memory");
}
__device__ __forceinline__ void wait_async0() {
  asm volatile("s_wait_asynccnt 0x0" ::: "memory");
}

// ---------------------------------------------------------------------------
// Fused GEMM + (optional) BN-stat partial reduction, fp32 WMMA 16x16x4,
// double-buffered LDS: A tile pipelined through registers (BN/ReLU fused at
// fetch), B tile (weights) copied with global_load_async_to_lds_b32.
// MODE 0: Y1 = z @ W1 + b1                         (K=128, N=256, stats out)
// MODE 1: Y2 = [relu(bn1(Y1)) | z] @ W2 + b2       (K=384, N=256, stats out)
// MODE 2: logits = [relu(bn2(Y2)) | relu(bn1(Y1)) | z] @ Wout + bout
//                                                  (K=640, N=366, no stats)
// ---------------------------------------------------------------------------
template <int MODE>
__global__ __launch_bounds__(256) void gemm_wmma(
    const float* __restrict__ z,   // [B,128]
    const float* __restrict__ Y1,  // [B,256] pre-BN layer-1 output
    const float* __restrict__ sc1, // [256] bn1 scale
    const float* __restrict__ sh1, // [256] bn1 shift
    const float* __restrict__ Y2,  // [B,256] pre-BN layer-2 output
    const float* __restrict__ sc2,
    const float* __restrict__ sh2,
    const float* __restrict__ W,    // [K,N] row-major
    const float* __restrict__ bias, // [N]
    float* __restrict__ Yout,       // [B,N]
    float* __restrict__ psum,       // [ROWBLK, 256] partial col sums
    float* __restrict__ psumsq)     // [ROWBLK, 256] partial col sum-of-squares
{
  constexpr int K   = (MODE == 0) ? 128 : (MODE == 1) ? 384 : 640;
  constexpr int N   = (MODE == 2) ? DATA_DIM : 256;
  constexpr int NKB = K / KT;

  __shared__ float As[2][MT][KT + 1]; // +1 pad: conflict-free M-strided reads
  __shared__ float Bs[2][KT][NT];
  __shared__ float Rs[16][NT]; // deterministic stat staging: 16 contributions/col
  __shared__ float Rq[16][NT];

  const int tid  = threadIdx.x;
  const int lane = tid & 31;
  const int wave = tid >> 5;   // 8 waves
  const int half = lane >> 4;  // 0/1: lane group
  const int l15  = lane & 15;
  const int wr   = wave & 3;   // 4 row-waves  (32 rows each)
  const int wc   = wave >> 2;  // 2 col-waves  (32 cols each)
  const int rowBase = blockIdx.x * MT;
  const int nBase   = blockIdx.y * NT;

  // --- fetch next A tile into registers, applying concat + BN-ReLU ---
  auto fetchA = [&](int kb, float* r) {
#pragma unroll
    for (int i = 0; i < 8; ++i) {
      int idx = tid * 8 + i;
      int row = rowBase + idx / KT;
      int c = kb + (idx % KT);
      float v;
      if (MODE == 0) {
        v = z[row * 128 + c];
      } else if (MODE == 1) {
        if (c < 256)
          v = fmaxf(Y1[row * 256 + c] * sc1[c] + sh1[c], 0.0f);
        else
          v = z[row * 128 + (c - 256)];
      } else {
        if (c < 256)
          v = fmaxf(Y2[row * 256 + c] * sc2[c] + sh2[c], 0.0f);
        else if (c < 512)
          v = fmaxf(Y1[row * 256 + (c - 256)] * sc1[c - 256] + sh1[c - 256], 0.0f);
        else
          v = z[row * 128 + (c - 512)];
      }
      r[i] = v;
    }
  };
  auto storeA = [&](int buf, const float* r) {
#pragma unroll
    for (int i = 0; i < 8; ++i) {
      int idx = tid * 8 + i;
      As[buf][idx / KT][idx % KT] = r[i];
    }
  };
  // --- async copy next B tile (weights) straight into LDS ---
  auto issueB = [&](int kb, int buf) {
#pragma unroll
    for (int i = 0; i < 4; ++i) {
      int idx = tid * 4 + i;
      int k = idx >> 6, n = idx & 63;
      int col = nBase + n;
      if (MODE != 2 || col < DATA_DIM) {
        unsigned laddr = (unsigned)(size_t)&Bs[buf][k][n];
        async_load_b32(laddr, &W[(size_t)(kb + k) * N + col]);
      }
    }
  };

  // pre-zero padded B columns once (masked lanes never overwrite them)
  if (MODE == 2) {
    for (int i = tid; i < 2 * KT * NT; i += 256) {
      int buf = i / (KT * NT), r = i % (KT * NT);
      int k = r / NT, n = r % NT;
      if (nBase + n >= DATA_DIM) Bs[buf][k][n] = 0.0f;
    }
  }

  v8f acc[2][2] = {}; // [mi][ni] 16x16 f32 accumulators
  float aNext[8];

  // prologue: stage tile 0
  fetchA(0, aNext);
  issueB(0, 0);
  storeA(0, aNext);
  wait_async0();
  __syncthreads();

  for (int kb = 0; kb < NKB; ++kb) {
    const int cur = kb & 1;
    const bool more = (kb + 1 < NKB);
    if (more) { // issue next tile's traffic before touching this tile
      fetchA(kb + 1, aNext);
      issueB(kb + 1, cur ^ 1);
    }

    // ---- 4 WMMA k-steps over the staged tile, 2x2 tiles per wave ----
#pragma unroll
    for (int kk = 0; kk < KT; kk += 4) {
      v2f a[2], b[2];
#pragma unroll
      for (int mi = 0; mi < 2; ++mi) {
        int m = wr * 32 + mi * 16 + l15;       // A: M = lane%16
        a[mi].x = As[cur][m][kk + 2 * half];   //    K = k0 + 2*half + vgpr
        a[mi].y = As[cur][m][kk + 2 * half + 1];
      }
#pragma unroll
      for (int ni = 0; ni < 2; ++ni) {
        int n = wc * 32 + ni * 16 + l15;       // B: N = lane%16
        b[ni].x = Bs[cur][kk + 2 * half][n];   //    K = k0 + 2*half + vgpr
        b[ni].y = Bs[cur][kk + 2 * half + 1][n];
      }
#pragma unroll
      for (int mi = 0; mi < 2; ++mi)
#pragma unroll
        for (int ni = 0; ni < 2; ++ni)
          acc[mi][ni] = __builtin_amdgcn_wmma_f32_16x16x4_f32(
              false, a[mi], false, b[ni], (short)0, acc[mi][ni], false, false);
    }

    if (more) storeA(cur ^ 1, aNext); // ds stores into the other buffer
    wait_async0();                    // this wave's async B copies done
    __syncthreads();                  // all waves' tile (kb+1) staged
  }

  // ---- store C (+bias); gather per-column BN partial stats ----
#pragma unroll
  for (int mi = 0; mi < 2; ++mi) {
#pragma unroll
    for (int ni = 0; ni < 2; ++ni) {
      int colLocal = wc * 32 + ni * 16 + l15;
      int col = nBase + colLocal;
      float bv = (MODE == 2 && col >= DATA_DIM) ? 0.0f : bias[col];
      float s = 0.0f, q = 0.0f;
#pragma unroll
      for (int r = 0; r < 8; ++r) { // C/D: row = 8*half + vgpr, col = lane%16
        int row = rowBase + wr * 32 + mi * 16 + 8 * half + r;
        float v = acc[mi][ni][r] + bv;
        if (MODE == 2) {
          if (col < DATA_DIM) Yout[row * DATA_DIM + col] = v;
        } else {
          Yout[row * 256 + col] = v;
          s += v;
          q += v * v;
        }
      }
      if (MODE != 2) {
        int cid = wr * 4 + mi * 2 + half; // 16 unique contributions per column
        Rs[cid][colLocal] = s;
        Rq[cid][colLocal] = q;
      }
    }
  }
  if (MODE != 2) {
    __syncthreads();
    if (tid < NT) { // fixed-order reduction -> deterministic across replays
      float s = 0.0f, q = 0.0f;
#pragma unroll
      for (int c = 0; c < 16; ++c) {
        s += Rs[c][tid];
        q += Rq[c][tid];
      }
      psum[blockIdx.x * 256 + nBase + tid] = s;
      psumsq[blockIdx.x * 256 + nBase + tid] = q;
    }
  }
}

// ---------------------------------------------------------------------------
// Finalize batch-norm: mu/var from partials -> fused (scale, shift).
// ---------------------------------------------------------------------------
__global__ __launch_bounds__(256) void bn_finalize(
    const float* __restrict__ psum, const float* __restrict__ psumsq,
    const float* __restrict__ gamma, const float* __restrict__ beta,
    float* __restrict__ scale, float* __restrict__ shift)
{
  int n = threadIdx.x; // 256 columns
  float s = 0.0f, q = 0.0f;
  for (int i = 0; i < ROWBLK; ++i) {
    s += psum[i * 256 + n];
    q += psumsq[i * 256 + n];
  }
  const float invB = 1.0f / (float)BATCH_N;
  float mu  = s * invB;
  float var = q * invB - mu * mu;
  float sc  = gamma[n] * rsqrtf(var + BN_EPS);
  scale[n] = sc;
  shift[n] = beta[n] - mu * sc;
}

// ---------------------------------------------------------------------------
// Ragged per-segment gumbel-softmax + tanh on alpha columns, in place on out.
// One block per row; segment table is a compile-time constant.
// ---------------------------------------------------------------------------
__global__ __launch_bounds__(384) void gumbel_rows(
    float* __restrict__ out, const float* __restrict__ g)
{
  __shared__ float s_logit[DATA_DIM];
  __shared__ float s_t[DATA_DIM];
  __shared__ float s_m[NSEG];
  __shared__ float s_inv[NSEG];

  const int row = blockIdx.x;
  const int tid = threadIdx.x;

  if (tid < DATA_DIM) {
    float lg = out[row * DATA_DIM + tid];
    s_logit[tid] = lg;
    s_t[tid] = (lg + g[row * DATA_DIM + tid]) * TAU_INV;
  }
  __syncthreads();

  if (tid < NSEG) { // one leader per segment (max span 100)
    int a = c_segStart[tid], b = c_segStart[tid + 1];
    float m = -__builtin_inff();
    for (int j = a; j < b; ++j) m = fmaxf(m, s_t[j]);
    float sum = 0.0f;
    for (int j = a; j < b; ++j) sum += __expf(s_t[j] - m);
    s_m[tid] = m;
    s_inv[tid] = 1.0f / sum;
  }
  __syncthreads();

  if (tid < DATA_DIM) {
    int seg = 0;
    for (int s = 1; s < NSEG; ++s)
      if (tid >= c_segStart[s]) seg = s;
    bool isAlpha = (seg < 20) && ((seg & 1) == 0);
    float v = isAlpha ? tanhf(s_logit[tid])
                      : __expf(s_t[tid] - s_m[seg]) * s_inv[seg];
    out[row * DATA_DIM + tid] = v;
  }
}

// ---------------------------------------------------------------------------
extern "C" void kernel_launch(void* const* d_in, const int* in_sizes, int n_in,
                              void* d_out, int out_size, void* d_ws, size_t ws_size,
                              hipStream_t stream) {
  (void)in_sizes; (void)n_in; (void)out_size; (void)ws_size;
  const float* z      = (const float*)d_in[0];
  const float* g      = (const float*)d_in[1];
  const float* W1     = (const float*)d_in[2];
  const float* b1     = (const float*)d_in[3];
  const float* gamma1 = (const float*)d_in[4];
  const float* beta1  = (const float*)d_in[5];
  const float* W2     = (const float*)d_in[6];
  const float* b2     = (const float*)d_in[7];
  const float* gamma2 = (const float*)d_in[8];
  const float* beta2  = (const float*)d_in[9];
  const float* Wout   = (const float*)d_in[10];
  const float* bout   = (const float*)d_in[11];
  float* out = (float*)d_out;

  // workspace carve-up (floats)
  float* ws = (float*)d_ws;
  size_t off = 0;
  float* Y1     = ws + off; off += (size_t)BATCH_N * 256; // 67 MB
  float* Y2     = ws + off; off += (size_t)BATCH_N * 256; // 67 MB
  float* psum1  = ws + off; off += (size_t)ROWBLK * 256;
  float* psq1   = ws + off; off += (size_t)ROWBLK * 256;
  float* psum2  = ws + off; off += (size_t)ROWBLK * 256;
  float* psq2   = ws + off; off += (size_t)ROWBLK * 256;
  float* scale1 = ws + off; off += 256;
  float* shift1 = ws + off; off += 256;
  float* scale2 = ws + off; off += 256;
  float* shift2 = ws + off; off += 256;

  dim3 blk(256);
  // layer 1: Y1 = z @ W1 + b1 (+ stats)
  gemm_wmma<0><<<dim3(ROWBLK, 256 / NT), blk, 0, stream>>>(
      z, nullptr, nullptr, nullptr, nullptr, nullptr, nullptr,
      W1, b1, Y1, psum1, psq1);
  bn_finalize<<<1, 256, 0, stream>>>(psum1, psq1, gamma1, beta1, scale1, shift1);

  // layer 2: Y2 = [relu(bn1(Y1)) | z] @ W2 + b2 (+ stats)
  gemm_wmma<1><<<dim3(ROWBLK, 256 / NT), blk, 0, stream>>>(
      z, Y1, scale1, shift1, nullptr, nullptr, nullptr,
      W2, b2, Y2, psum2, psq2);
  bn_finalize<<<1, 256, 0, stream>>>(psum2, psq2, gamma2, beta2, scale2, shift2);

  // output layer: logits -> d_out (366 cols, padded to 384 in tiling)
  gemm_wmma<2><<<dim3(ROWBLK, 384 / NT), blk, 0, stream>>>(
      z, Y1, scale1, shift1, Y2, scale2, shift2,
      Wout, bout, out, nullptr, nullptr);

  // ragged gumbel-softmax / tanh, in place on d_out
  gumbel_rows<<<BATCH_N, 384, 0, stream>>>(out, g);
}